// VanillaRNN_75265006895137
// MI455X (gfx1250) — compile-verified
//
#include <hip/hip_runtime.h>
#include <math.h>
#include <stdint.h>

// Problem constants (from reference)
static constexpr int kV = 32000;
static constexpr int kE = 512;
static constexpr int kH = 1024;
static constexpr int kB = 16;
static constexpr int kS = 256;

// TDM pads 1 DWORD per 256 DWORDs of stream => element (r,c) of a [rows x cols]
// tile lives at dword index r*pitch + c + (c>>8), pitch = cols + cols/256.
static constexpr int kEP = kE + kE / 256;   // 514
static constexpr int kHP = kH + kH / 256;   // 1028

static constexpr int kNBLK = 64;   // persistent scan workgroups (one 16-col slice each)
static constexpr int kRT   = 128;  // scan block: 4 waves

// LDS float offsets inside rnn_scan's dynamic LDS
static constexpr int kOffWih0 = 0;                        // 16*514
static constexpr int kOffWhh0 = kOffWih0 + 16 * kEP;      // 8224
static constexpr int kOffWih1 = kOffWhh0 + 16 * kHP;      // 24672
static constexpr int kOffWhh1 = kOffWih1 + 16 * kHP;      // 41120
static constexpr int kOffRed  = kOffWhh1 + 16 * kHP;      // 57568
static constexpr int kScanLds = kOffRed + 4 * 256;        // 58592 floats (~229 KB)

typedef __attribute__((ext_vector_type(2))) float v2f;
typedef __attribute__((ext_vector_type(8))) float v8f;
typedef __attribute__((ext_vector_type(4))) unsigned int u32x4;
typedef __attribute__((ext_vector_type(8))) int i32x8;
typedef __attribute__((ext_vector_type(4))) int i32x4;

#define WMMA4(a, b, c) \
  __builtin_amdgcn_wmma_f32_16x16x4_f32(false, (a), false, (b), (short)0, (c), false, false)

__device__ inline int lidx(int r, int c, int pitch) { return r * pitch + c + (c >> 8); }

// ---------------------------------------------------------------------------
// TDM: DMA a [rows x cols] fp32 tile (row stride = stride elems) from global
// into LDS at byte offset lds_off, with 1-dword padding per 256 dwords.
// Descriptor bit layout per CDNA5 ISA ch.8 (groups 0/1; 2D so groups 2/3 = 0).
// This toolchain exposes the 6-arg builtin (g0, g1, g2, g3, extra, cpol).
// ---------------------------------------------------------------------------
__device__ inline void tdm_load_2d(const void* gptr, unsigned lds_off,
                                   unsigned rows, unsigned cols, unsigned stride) {
  unsigned long long ga = (unsigned long long)(uintptr_t)gptr;
  u32x4 g0;
  g0[0] = 1u;                                            // count=1 (valid user D#)
  g0[1] = lds_off;                                       // lds_addr (bytes)
  g0[2] = (unsigned)(ga & 0xffffffffu);                  // global_addr[31:0]
  g0[3] = (unsigned)((ga >> 32) & 0x01ffffffu)           // global_addr[56:32]
          | 0x80000000u;                                 // type=2 ("image")
  i32x8 g1;
  g1[0] = (int)((2u << 16)     // data_size = 4 bytes
                | (1u << 20)   // pad_enable
                | (7u << 22)); // pad_interval = 256 dwords (pad_amount=0 => 1 dword)
  g1[1] = (int)((cols & 0xffffu) << 16);                     // tensor_dim0[15:0]
  g1[2] = (int)(((cols >> 16) & 0xffffu) |                   // tensor_dim0[31:16]
                ((rows & 0xffffu) << 16));                   // tensor_dim1[15:0]
  g1[3] = (int)(((rows >> 16) & 0xffffu) |                   // tensor_dim1[31:16]
                ((cols & 0xffffu) << 16));                   // tile_dim0
  g1[4] = (int)(rows & 0xffffu);                             // tile_dim1 (tile_dim2=0)
  g1[5] = (int)stride;                                       // tensor_dim0_stride[31:0]
  g1[6] = 0;                                                 // stride[47:32], dim1_stride lo
  g1[7] = 0;
  i32x4 z4 = {0, 0, 0, 0};
  i32x8 z8 = {0, 0, 0, 0, 0, 0, 0, 0};
  __builtin_amdgcn_tensor_load_to_lds(g0, g1, z4, z4, z8, 0);
}

// ---------------------------------------------------------------------------
// init: zero h ping-pong state + global barrier counter (fresh every call)
// ---------------------------------------------------------------------------
__global__ void rnn_init(float* h0buf, float* h1buf, unsigned* bar) {
  int i = blockIdx.x * blockDim.x + threadIdx.x;
  if (i < 2 * kB * kH) {
    h0buf[i] = 0.0f;
    h1buf[i] = 0.0f;
  }
  if (i == 0) *bar = 0u;
}

// ---------------------------------------------------------------------------
// device-wide barrier: monotonic counter, per-instance target (no reset race)
// ---------------------------------------------------------------------------
__device__ inline void grid_barrier(unsigned* bar, unsigned target) {
  __syncthreads();
  __threadfence();
  if (threadIdx.x == 0) {
    atomicAdd(bar, 1u);
    while (__hip_atomic_load(bar, __ATOMIC_RELAXED, __HIP_MEMORY_SCOPE_AGENT) < target) {
      __builtin_amdgcn_s_sleep(1);
    }
  }
  __syncthreads();
  __threadfence();
}

// dual-accumulator K-slab: A rows from global, B rows from (TDM-padded) LDS
__device__ inline void gemm_slab(const float* __restrict__ arow,
                                 const float* __restrict__ sW, int pitch, int ml,
                                 int koff, int kbeg, int kend, v8f& acc0, v8f& acc1) {
#pragma unroll 4
  for (int k0 = kbeg; k0 < kend; k0 += 8) {
    int ka = k0 + koff, kb = k0 + 4 + koff;
    v2f a0 = *(const v2f*)(arow + ka);
    v2f a1 = *(const v2f*)(arow + kb);
    v2f b0, b1;
    int ia = lidx(ml, ka, pitch), ib = lidx(ml, kb, pitch);
    b0[0] = sW[ia]; b0[1] = sW[ia + 1];
    b1[0] = sW[ib]; b1[1] = sW[ib + 1];
    acc0 = WMMA4(a0, b0, acc0);
    acc1 = WMMA4(a1, b1, acc1);
  }
}

// ---------------------------------------------------------------------------
// Persistent 2-layer RNN scan. grid = 64 blocks x 128 threads.
// Block wg owns output columns [wg*16, wg*16+16) of both layers; its weight
// slices are DMA'd once into LDS (~229 KB of the 320 KB/WGP) by the TDM.
// ---------------------------------------------------------------------------
__global__ void __launch_bounds__(kRT, 1)
rnn_scan(const int* __restrict__ x, const float* __restrict__ emb,
         const float* __restrict__ Wih0, const float* __restrict__ Whh0,
         const float* __restrict__ bih0, const float* __restrict__ bhh0,
         const float* __restrict__ Wih1, const float* __restrict__ Whh1,
         const float* __restrict__ bih1, const float* __restrict__ bhh1,
         float* __restrict__ h0buf, float* __restrict__ h1buf,
         float* __restrict__ oseq, float* __restrict__ hidden,
         unsigned* __restrict__ bar) {
  extern __shared__ float lds[];
  const float* sWih0 = lds + kOffWih0;
  const float* sWhh0 = lds + kOffWhh0;
  const float* sWih1 = lds + kOffWih1;
  const float* sWhh1 = lds + kOffWhh1;
  float*       sred  = lds + kOffRed;   // [4][256] partial C tiles

  const int tid  = threadIdx.x;
  const int wave = tid >> 5;
  const int lane = tid & 31;
  const int nb   = blockIdx.x << 4;     // column base of this WG's slice
  const int ml   = lane & 15;           // A-row (batch) / B-row (out col) index
  const int koff = (lane >> 4) << 1;    // 0 for lanes 0-15, 2 for lanes 16-31

  // Stage this WG's four weight slices via the Tensor Data Mover
  if (wave == 0) {
    const unsigned base = __builtin_amdgcn_groupstaticsize();
    tdm_load_2d(Wih0 + (size_t)nb * kE, base + 4u * kOffWih0, 16, kE, kE);
    tdm_load_2d(Whh0 + (size_t)nb * kH, base + 4u * kOffWhh0, 16, kH, kH);
    tdm_load_2d(Wih1 + (size_t)nb * kH, base + 4u * kOffWih1, 16, kH, kH);
    tdm_load_2d(Whh1 + (size_t)nb * kH, base + 4u * kOffWhh1, 16, kH, kH);
    __builtin_amdgcn_s_wait_tensorcnt(0);
  }
  __syncthreads();

  const v8f vzero = {};
  unsigned bt = 0;

  // ---- prologue: h0(0) = tanh(emb(x[:,0]) @ Wih0^T + b)   (h0(-1)==0)
  {
    const int xi = x[ml * kS + 0];
    const float* erow = emb + (size_t)xi * kE;
    v8f acc0 = vzero, acc1 = vzero;
    gemm_slab(erow, sWih0, kEP, ml, koff, wave * (kE / 4), wave * (kE / 4) + (kE / 4),
              acc0, acc1);
#pragma unroll
    for (int i = 0; i < 8; ++i) sred[wave * 256 + i * 32 + lane] = acc0[i] + acc1[i];
    __syncthreads();
    for (int e = tid; e < 256; e += kRT) {
      int mm = e >> 4, nn = e & 15;
      int si = ((mm & 7) << 5) + ((mm >> 3) << 4) + nn;
      float s = sred[si] + sred[256 + si] + sred[512 + si] + sred[768 + si];
      s += bih0[nb + nn] + bhh0[nb + nn];
      h0buf[mm * kH + nb + nn] = tanhf(s);  // slot 0 = h0(0)
    }
    bt += kNBLK;
    grid_barrier(bar, bt);
  }

  // ---- main loop: per t (h0(t), h1(t-1) published) compute h1(t), h0(t+1)
  for (int t = 0; t < kS; ++t) {
    const float* h0cur  = h0buf + (t & 1) * (kB * kH);
    const float* h1prev = h1buf + ((t + 1) & 1) * (kB * kH);
    float*       h1cur  = h1buf + (t & 1) * (kB * kH);
    float*       h0nxt  = h0buf + ((t + 1) & 1) * (kB * kH);

    // ---- layer 1: h1(t) = tanh(h0(t)@Wih1^T + h1(t-1)@Whh1^T + b)
    {
      v8f acc0 = vzero, acc1 = vzero;
      gemm_slab(h0cur + ml * kH, sWih1, kHP, ml, koff,
                wave * 256, wave * 256 + 256, acc0, acc1);
      gemm_slab(h1prev + ml * kH, sWhh1, kHP, ml, koff,
                wave * 256, wave * 256 + 256, acc0, acc1);
#pragma unroll
      for (int i = 0; i < 8; ++i) sred[wave * 256 + i * 32 + lane] = acc0[i] + acc1[i];
      __syncthreads();
      for (int e = tid; e < 256; e += kRT) {
        int mm = e >> 4, nn = e & 15;
        int si = ((mm & 7) << 5) + ((mm >> 3) << 4) + nn;
        float s = sred[si] + sred[256 + si] + sred[512 + si] + sred[768 + si];
        s += bih1[nb + nn] + bhh1[nb + nn];
        s = tanhf(s);
        h1cur[mm * kH + nb + nn] = s;
        oseq[((size_t)mm * kS + t) * kH + nb + nn] = s;
      }
      __syncthreads();
    }

    // ---- layer 0: h0(t+1) = tanh(emb(x[:,t+1])@Wih0^T + h0(t)@Whh0^T + b)
    if (t + 1 < kS) {
      const int xi = x[ml * kS + (t + 1)];
      const float* erow = emb + (size_t)xi * kE;
      v8f acc0 = vzero, acc1 = vzero;
      gemm_slab(erow, sWih0, kEP, ml, koff,
                wave * (kE / 4), wave * (kE / 4) + (kE / 4), acc0, acc1);
      gemm_slab(h0cur + ml * kH, sWhh0, kHP, ml, koff,
                wave * 256, wave * 256 + 256, acc0, acc1);
#pragma unroll
      for (int i = 0; i < 8; ++i) sred[wave * 256 + i * 32 + lane] = acc0[i] + acc1[i];
      __syncthreads();
      for (int e = tid; e < 256; e += kRT) {
        int mm = e >> 4, nn = e & 15;
        int si = ((mm & 7) << 5) + ((mm >> 3) << 4) + nn;
        float s = sred[si] + sred[256 + si] + sred[512 + si] + sred[768 + si];
        s += bih0[nb + nn] + bhh0[nb + nn];
        h0nxt[mm * kH + nb + nn] = tanhf(s);
      }
    }
    bt += kNBLK;
    grid_barrier(bar, bt);
  }

  // ---- epilogue: final hidden states [2][B][H] (this WG's own columns)
  const float* h0f = h0buf + ((kS - 1) & 1) * (kB * kH);
  const float* h1f = h1buf + ((kS - 1) & 1) * (kB * kH);
  for (int i = tid; i < 256; i += kRT) {
    int r = i >> 4, c = i & 15;
    hidden[(size_t)r * kH + nb + c] = h0f[r * kH + nb + c];
    hidden[(size_t)kB * kH + (size_t)r * kH + nb + c] = h1f[r * kH + nb + c];
  }
}

// ---------------------------------------------------------------------------
// Output projection: logits[m, n] = oseq[m, :] . Wout[n, :] + bout[n]
// M = B*S = 4096, N = V = 32000, K = H = 1024. HBM-write-bound -> fp32 WMMA.
// Block: 256 threads = 8 waves; wave owns a 16x16 tile; A tile DMA'd by TDM.
// ---------------------------------------------------------------------------
__global__ void __launch_bounds__(256)
out_proj(const float* __restrict__ oseq, const float* __restrict__ Wout,
         const float* __restrict__ bout, float* __restrict__ logits) {
  extern __shared__ float sA[];  // [16][kHP] (TDM-padded)
  const int tid  = threadIdx.x;
  const int wave = tid >> 5;
  const int lane = tid & 31;
  const int mb = blockIdx.y << 4;                 // row tile base (16 rows)
  const int n  = blockIdx.x * 128 + wave * 16 + (lane & 15);
  const int ml = lane & 15;
  const int koff = (lane >> 4) << 1;

  if (wave == 0) {
    tdm_load_2d(oseq + (size_t)mb * kH, __builtin_amdgcn_groupstaticsize(), 16, kH, kH);
    __builtin_amdgcn_s_wait_tensorcnt(0);
  }
  __syncthreads();

  const float* wr = Wout + (size_t)n * kH;
  v8f acc0 = {}, acc1 = {};
#pragma unroll 4
  for (int k0 = 0; k0 < kH; k0 += 8) {
    int ka = k0 + koff, kb = k0 + 4 + koff;
    v2f a0, a1;
    int ia = lidx(ml, ka, kHP), ib = lidx(ml, kb, kHP);
    a0[0] = sA[ia]; a0[1] = sA[ia + 1];
    a1[0] = sA[ib]; a1[1] = sA[ib + 1];
    v2f b0 = *(const v2f*)(wr + ka);
    v2f b1 = *(const v2f*)(wr + kb);
    acc0 = WMMA4(a0, b0, acc0);
    acc1 = WMMA4(a1, b1, acc1);
  }

  const float bias = bout[n];
  const int mhalf = (lane >> 4) << 3;
#pragma unroll
  for (int i = 0; i < 8; ++i) {
    logits[(size_t)(mb + i + mhalf) * kV + n] = acc0[i] + acc1[i] + bias;
  }
}

// ---------------------------------------------------------------------------
extern "C" void kernel_launch(void* const* d_in, const int* in_sizes, int n_in,
                              void* d_out, int out_size, void* d_ws, size_t ws_size,
                              hipStream_t stream) {
  const int*   x    = (const int*)d_in[0];
  const float* emb  = (const float*)d_in[1];
  const float* Wih0 = (const float*)d_in[2];
  const float* Whh0 = (const float*)d_in[3];
  const float* bih0 = (const float*)d_in[4];
  const float* bhh0 = (const float*)d_in[5];
  const float* Wih1 = (const float*)d_in[6];
  const float* Whh1 = (const float*)d_in[7];
  const float* bih1 = (const float*)d_in[8];
  const float* bhh1 = (const float*)d_in[9];
  const float* Wout = (const float*)d_in[10];
  const float* bout = (const float*)d_in[11];

  // Workspace: h ping-pongs (2x32K floats) + out_seq (4M floats) + barrier
  float* ws    = (float*)d_ws;
  float* h0buf = ws;                                  // 2*B*H
  float* h1buf = ws + 2 * kB * kH;                    // 2*B*H
  float* oseq  = ws + 4 * kB * kH;                    // B*S*H
  unsigned* bar = (unsigned*)(ws + 4 * kB * kH + (size_t)kB * kS * kH);

  float* logits = (float*)d_out;                      // [B][S][V]
  float* hidden = logits + (size_t)kB * kS * kV;      // [2][B][H]

  rnn_init<<<dim3(128), dim3(256), 0, stream>>>(h0buf, h1buf, bar);

  const size_t lds_scan = (size_t)kScanLds * sizeof(float);  // ~229 KB
  (void)hipFuncSetAttribute((const void*)rnn_scan,
                            hipFuncAttributeMaxDynamicSharedMemorySize, (int)lds_scan);
  rnn_scan<<<dim3(kNBLK), dim3(kRT), lds_scan, stream>>>(
      x, emb, Wih0, Whh0, bih0, bhh0, Wih1, Whh1, bih1, bhh1,
      h0buf, h1buf, oseq, hidden, bar);

  const size_t lds_proj = (size_t)(16 * kHP) * sizeof(float);  // ~64 KB
  (void)hipFuncSetAttribute((const void*)out_proj,
                            hipFuncAttributeMaxDynamicSharedMemorySize, (int)lds_proj);
  out_proj<<<dim3(kV / 128, (kB * kS) / 16), dim3(256), lds_proj, stream>>>(
      oseq, Wout, bout, logits);
}